// Context_63488206570149
// MI455X (gfx1250) — compile-verified
//
#include <hip/hip_runtime.h>
#include <math.h>

typedef __attribute__((ext_vector_type(2))) float v2f;
typedef __attribute__((ext_vector_type(4))) float v4f;
typedef __attribute__((ext_vector_type(8))) float v8f;

#define H      512
#define NSEG   64
#define NROWS  262144
#define SEG_ROWS_PER_BLOCK 256

// d_ws layout (floats):
//  [0,       32768)  seg_sum [64][512]
//  [32768,   32896)  counts  [64] (+pad)
//  [32896,   65664)  hdn     [64][512]
//  [65664,   98432)  gate    [64][512]
#define WS_SEGSUM 0
#define WS_COUNTS 32768
#define WS_HDN    32896
#define WS_GATE   65664
#define WS_ZERO_N 32896   // seg_sum + counts region to clear

__global__ void zero_kernel(float* __restrict__ p, int n) {
    int i = blockIdx.x * blockDim.x + threadIdx.x;
    if (i < n) p[i] = 0.0f;
}

// Block: 128 threads, each owning 4 contiguous columns (float4 = b128 loads),
// so one block covers the full 512-column row. Each block walks 256
// consecutive rows. batch_id is sorted, so run-length accumulate in registers
// and flush with atomicAdds only on segment transitions (uniform branch).
// h_V is streamed exactly once -> non-temporal loads keep it from displacing
// the small tables in L2.
__global__ void seg_sum_kernel(const float* __restrict__ hV,
                               const int*   __restrict__ bid,
                               float* __restrict__ seg_sum,
                               float* __restrict__ counts) {
    const int c4   = threadIdx.x;             // float4 column index, 0..127
    const int row0 = blockIdx.x * SEG_ROWS_PER_BLOCK;
    const bool doCount = (threadIdx.x == 0);

    const v4f* __restrict__ hv4 = (const v4f*)hV;

    int   cur      = bid[row0];
    v4f   acc      = {};
    float runCount = 0.0f;

    for (int r = 0; r < SEG_ROWS_PER_BLOCK; ++r) {
        const int row = row0 + r;
        const int seg = bid[row];             // uniform per block -> scalar load
        if (seg != cur) {                     // uniform branch
            float* dst = &seg_sum[cur * H + c4 * 4];
            atomicAdd(dst + 0, acc.x);
            atomicAdd(dst + 1, acc.y);
            atomicAdd(dst + 2, acc.z);
            atomicAdd(dst + 3, acc.w);
            acc = (v4f){};
            if (doCount) atomicAdd(&counts[cur], runCount);
            runCount = 0.0f;
            cur = seg;
        }
        v4f h = __builtin_nontemporal_load(&hv4[(size_t)row * (H / 4) + c4]);
        acc.x += h.x; acc.y += h.y; acc.z += h.z; acc.w += h.w;
        runCount += 1.0f;
    }
    {
        float* dst = &seg_sum[cur * H + c4 * 4];
        atomicAdd(dst + 0, acc.x);
        atomicAdd(dst + 1, acc.y);
        atomicAdd(dst + 2, acc.z);
        atomicAdd(dst + 3, acc.w);
        if (doCount) atomicAdd(&counts[cur], runCount);
    }
}

// One wave (32 threads) computes one 16x16 output tile of  A @ W^T + bias.
// ACT==0: A = seg_sum scaled by 1/max(count,1)   (i.e. c_V), relu epilogue.
// ACT==1: A = hdn, sigmoid epilogue.
//
// V_WMMA_F32_16X16X4_F32 operand layout (ISA 7.12.2, 32-bit A 16x4):
//   lane L<16 : M=L,    holds K = k0, k0+1   -> float2 at A[(s0+L)*H   + k0]
//   lane L>=16: M=L-16, holds K = k0+2, k0+3 -> float2 at A[(s0+L-16)*H + k0+2]
// B (4x16) mirrors with lane = output column n; since out[s][j] = sum_k A[s][k]*W[j][k],
// B[k][n] = W[j0+n][k] and the same float2 addressing applies to W rows.
template <int ACT>
__global__ void gemm_wmma_kernel(const float* __restrict__ A,
                                 const float* __restrict__ counts,
                                 const float* __restrict__ W,
                                 const float* __restrict__ bias,
                                 float* __restrict__ out) {
    const int lane = threadIdx.x;        // 0..31
    const int j0   = blockIdx.x * 16;    // output column tile
    const int s0   = blockIdx.y * 16;    // output row tile
    const int m    = lane & 15;
    const int koff = (lane >> 4) * 2;

    float ainv = 1.0f;
    if (ACT == 0) ainv = 1.0f / fmaxf(counts[s0 + m], 1.0f);

    const float* Arow = A + (size_t)(s0 + m) * H + koff;
    const float* Brow = W + (size_t)(j0 + m) * H + koff;

    v8f acc = {};
#pragma unroll 4
    for (int k0 = 0; k0 < H; k0 += 4) {
        v2f a = *(const v2f*)(Arow + k0);
        if (ACT == 0) { a.x *= ainv; a.y *= ainv; }
        v2f b = *(const v2f*)(Brow + k0);
        // (neg_a, A, neg_b, B, c_mod, C, reuse_a, reuse_b)
        acc = __builtin_amdgcn_wmma_f32_16x16x4_f32(
            false, a, false, b, (short)0, acc, false, false);
    }

    // C/D layout: VGPR vr, lanes 0-15 -> row s0+vr, lanes 16-31 -> row s0+8+vr; col = j0+(lane&15)
    const int   ocol = j0 + m;
    const int   orow = s0 + (lane >> 4) * 8;
    const float bv   = bias[ocol];
#pragma unroll
    for (int vr = 0; vr < 8; ++vr) {
        float x = acc[vr] + bv;
        if (ACT == 0) x = fmaxf(x, 0.0f);
        else          x = 1.0f / (1.0f + __expf(-x));
        out[(size_t)(orow + vr) * H + ocol] = x;
    }
}

// out[i][j] = h_V[i][j] * gate[batch_id[i]][j], float4-vectorized.
// h_V / out are pure streams (NT hints); gate (128 KB) and batch_id (1 MB)
// stay resident in L2 at regular temporal priority.
__global__ void gate_kernel(const float* __restrict__ hV,
                            const int*   __restrict__ bid,
                            const float* __restrict__ gate,
                            float* __restrict__ out) {
    const size_t idx = (size_t)blockIdx.x * blockDim.x + threadIdx.x;
    const int row = (int)(idx >> 7);     // H/4 = 128 float4 per row
    const int c4  = (int)(idx & 127);
    const int seg = bid[row];

    const v4f h = __builtin_nontemporal_load(&((const v4f*)hV)[idx]);
    const v4f g = ((const v4f*)gate)[(size_t)seg * 128 + c4];
    v4f o;
    o.x = h.x * g.x; o.y = h.y * g.y; o.z = h.z * g.z; o.w = h.w * g.w;
    __builtin_nontemporal_store(o, &((v4f*)out)[idx]);
}

extern "C" void kernel_launch(void* const* d_in, const int* in_sizes, int n_in,
                              void* d_out, int out_size, void* d_ws, size_t ws_size,
                              hipStream_t stream) {
    const float* hV  = (const float*)d_in[0];
    const int*   bid = (const int*)  d_in[1];
    const float* W1  = (const float*)d_in[2];
    const float* b1  = (const float*)d_in[3];
    const float* W2  = (const float*)d_in[4];
    const float* b2  = (const float*)d_in[5];
    float* out = (float*)d_out;
    float* ws  = (float*)d_ws;

    float* seg_sum = ws + WS_SEGSUM;
    float* counts  = ws + WS_COUNTS;
    float* hdn     = ws + WS_HDN;
    float* gate    = ws + WS_GATE;

    // 1) clear accumulation scratch
    zero_kernel<<<(WS_ZERO_N + 255) / 256, 256, 0, stream>>>(ws, WS_ZERO_N);

    // 2) segment sums + counts
    seg_sum_kernel<<<NROWS / SEG_ROWS_PER_BLOCK, 128, 0, stream>>>(hV, bid, seg_sum, counts);

    // 3) hdn = relu(c_V @ W1^T + b1)   (c_V formed on the fly from seg_sum/counts)
    dim3 g2(H / 16, NSEG / 16);
    gemm_wmma_kernel<0><<<g2, 32, 0, stream>>>(seg_sum, counts, W1, b1, hdn);

    // 4) gate = sigmoid(hdn @ W2^T + b2)
    gemm_wmma_kernel<1><<<g2, 32, 0, stream>>>(hdn, counts, W2, b2, gate);

    // 5) out = h_V * gate[batch_id]
    const size_t total4 = (size_t)NROWS * H / 4;
    gate_kernel<<<(unsigned)(total4 / 256), 256, 0, stream>>>(hV, bid, gate, out);
}